// NormalizedChamferLoss_74861279969287
// MI455X (gfx1250) — compile-verified
//
#include <hip/hip_runtime.h>

// Chamfer loss on a fixed 96x96 grid, B=4.
// Distance tiles computed on the matrix pipe: with
//   a = (yq, xq, yq^2+xq^2, 1),  b = (-2yc, -2xc, 1, yc^2+xc^2 + BIG*(1-mask))
// a.b == masked squared distance, so one V_WMMA_F32_16X16X4_F32 = 256 masked d^2.

#define BATCH 4
#define HH 96
#define WW 96
#define NPTS (HH * WW)               // 9216
#define NTILES (NPTS / 16)           // 576
#define QT_PER_BLOCK 8               // 8 waves/block, one query tile each
#define BLOCKS_PER_BD (NTILES / QT_PER_BLOCK)  // 72
#define THRESH 0.5f
#define BIGF 1.0e9f

typedef __attribute__((ext_vector_type(2))) float v2f;
typedef __attribute__((ext_vector_type(8))) float v8f;

// ws layout (floats):
//   [0..3]  cntP per sample      [4..7] cntT per sample
//   [16 .. 16 + 8*NTILES) per-wave partial sums, idx = (b*2+dir)*NTILES + qtile

__global__ void chamfer_count_kernel(const float* __restrict__ pred,
                                     const float* __restrict__ tgt,
                                     float* __restrict__ ws) {
  __shared__ float sp[256];
  __shared__ float st[256];
  const int b = blockIdx.x;
  const int tid = threadIdx.x;
  const float* p = pred + b * NPTS;
  const float* t = tgt + b * NPTS;
  float cp = 0.0f, ct = 0.0f;
  for (int i = tid; i < NPTS; i += 256) {
    cp += (p[i] > THRESH) ? 1.0f : 0.0f;
    ct += (t[i] > THRESH) ? 1.0f : 0.0f;
  }
  sp[tid] = cp;
  st[tid] = ct;
  __syncthreads();
  for (int s = 128; s > 0; s >>= 1) {
    if (tid < s) { sp[tid] += sp[tid + s]; st[tid] += st[tid + s]; }
    __syncthreads();
  }
  if (tid == 0) {
    ws[b] = sp[0];
    ws[4 + b] = st[0];
  }
}

__global__ void chamfer_tile_kernel(const float* __restrict__ pred,
                                    const float* __restrict__ tgt,
                                    float* __restrict__ ws) {
  extern __shared__ float cdat[];  // NPTS * 4 floats = 144 KB (CDNA5 WGP LDS)
  const int tid = threadIdx.x;
  const int bd = blockIdx.x / BLOCKS_PER_BD;  // b*2 + dir
  const int qgroup = blockIdx.x % BLOCKS_PER_BD;
  const int dir = bd & 1;
  const int b = bd >> 1;
  const float* qbuf = (dir == 0) ? (pred + b * NPTS) : (tgt + b * NPTS);
  const float* cbuf = (dir == 0) ? (tgt + b * NPTS) : (pred + b * NPTS);
  const float invH = 1.0f / (float)(HH - 1);
  const float invW = 1.0f / (float)(WW - 1);

  // Build candidate B-operand table once per block into LDS:
  // per candidate: { -2y, -2x, 1, y^2+x^2 + BIG*(1-mask) }
  for (int c = tid; c < NPTS; c += 256) {
    float yc = (float)(c / WW) * invH;
    float xc = (float)(c % WW) * invW;
    float big = (cbuf[c] > THRESH) ? 0.0f : BIGF;
    cdat[c * 4 + 0] = -2.0f * yc;
    cdat[c * 4 + 1] = -2.0f * xc;
    cdat[c * 4 + 2] = 1.0f;
    cdat[c * 4 + 3] = yc * yc + xc * xc + big;
  }
  __syncthreads();

  const int lid = tid & 31;
  const int wave = tid >> 5;
  const int qtile = qgroup * QT_PER_BLOCK + wave;
  const int m = lid & 15;
  const int half = lid >> 4;

  // A operand: 16x4 f32 layout — lanes 0-15 hold K=0,1; lanes 16-31 hold K=2,3
  const int q = qtile * 16 + m;
  const float yq = (float)(q / WW) * invH;
  const float xq = (float)(q % WW) * invW;
  const float sqq = yq * yq + xq * xq;
  v2f A;
  A.x = half ? sqq : yq;   // K=2 : K=0
  A.y = half ? 1.0f : xq;  // K=3 : K=1
  const float qval = qbuf[q];

  v8f acc;
#pragma unroll
  for (int r = 0; r < 8; ++r) acc[r] = 4.0e9f;

  // B operand: 4x16 f32 layout mirrors A (lane = column, halves split K)
  const float* myc = cdat + (m * 4 + half * 2);
#pragma unroll 4
  for (int ct = 0; ct < NTILES; ++ct) {
    v2f Bv = *(const v2f*)(myc + ct * 64);  // ds_load_b64
    v8f c0 = {};
    v8f d = __builtin_amdgcn_wmma_f32_16x16x4_f32(
        false, A, false, Bv, (short)0, c0, false, false);
#pragma unroll
    for (int r = 0; r < 8; ++r) acc[r] = fminf(acc[r], d[r]);
  }

  // Row minima: butterfly min over the 16 lanes of each half.
  // After reduction every lane of a half holds rowmin for (M = r + 8*half).
  float psum = 0.0f;
#pragma unroll
  for (int r = 0; r < 8; ++r) {
    float v = acc[r];
    v = fminf(v, __shfl_xor(v, 1, 32));
    v = fminf(v, __shfl_xor(v, 2, 32));
    v = fminf(v, __shfl_xor(v, 4, 32));
    v = fminf(v, __shfl_xor(v, 8, 32));
    float dmin = sqrtf(fmaxf(v, 0.0f));
    float wgt = (qval > THRESH) ? 1.0f : 0.0f;
    // lane r owns row r (qval loaded at q=qtile*16+r); lane 24+r owns row 8+r
    bool sel = (lid == r) || (lid == 24 + r);
    psum += sel ? (wgt * dmin) : 0.0f;
  }

  // wave-wide sum of the 16 selected row contributions
  psum += __shfl_xor(psum, 16, 32);
  psum += __shfl_xor(psum, 8, 32);
  psum += __shfl_xor(psum, 4, 32);
  psum += __shfl_xor(psum, 2, 32);
  psum += __shfl_xor(psum, 1, 32);
  if (lid == 0) ws[16 + bd * NTILES + qtile] = psum;
}

__global__ void chamfer_final_kernel(const float* __restrict__ ws,
                                     float* __restrict__ out) {
  __shared__ float red[256];
  __shared__ float csum[8];
  const int tid = threadIdx.x;
  for (int combo = 0; combo < 8; ++combo) {
    float local = 0.0f;
    for (int i = tid; i < NTILES; i += 256)
      local += ws[16 + combo * NTILES + i];
    red[tid] = local;
    __syncthreads();
    for (int s = 128; s > 0; s >>= 1) {
      if (tid < s) red[tid] += red[tid + s];
      __syncthreads();
    }
    if (tid == 0) csum[combo] = red[0];
    __syncthreads();
  }
  if (tid == 0) {
    float total = 0.0f;
    for (int b = 0; b < BATCH; ++b) {
      float np = ws[b], nt = ws[4 + b];
      float mp = csum[b * 2 + 0] / fmaxf(np, 1.0f);
      float mt = csum[b * 2 + 1] / fmaxf(nt, 1.0f);
      total += ((np > 0.0f) && (nt > 0.0f)) ? (mp + mt) : 0.0f;
    }
    out[0] = total / (float)BATCH;
  }
}

extern "C" void kernel_launch(void* const* d_in, const int* in_sizes, int n_in,
                              void* d_out, int out_size, void* d_ws, size_t ws_size,
                              hipStream_t stream) {
  const float* pred = (const float*)d_in[0];
  const float* tgt = (const float*)d_in[1];
  float* out = (float*)d_out;
  float* ws = (float*)d_ws;

  chamfer_count_kernel<<<BATCH, 256, 0, stream>>>(pred, tgt, ws);
  chamfer_tile_kernel<<<BATCH * 2 * BLOCKS_PER_BD, 256,
                        NPTS * 4 * sizeof(float), stream>>>(pred, tgt, ws);
  chamfer_final_kernel<<<1, 256, 0, stream>>>(ws, out);
}